// GCNConv_4140348474047
// MI455X (gfx1250) — compile-verified
//
#include <hip/hip_runtime.h>
#include <math.h>
#include <cstdint>

// ---------------------------------------------------------------------------
// GCNConv fused pipeline for MI455X (gfx1250, wave32)
//   h   = GELU( BN( x @ W^T + b ) )        -> d_ws   (f32, exact via WMMA f32)
//   out = scatter_add( h[src] * norm, dst) -> d_out  (f32 hardware atomics)
// ---------------------------------------------------------------------------

typedef __attribute__((ext_vector_type(2))) float v2f;
typedef __attribute__((ext_vector_type(8))) float v8f;

#define B_SZ   4
#define N_SZ   10000
#define E_SZ   320000
#define D_IN   128
#define D_OUT  128
#define LDS_W_STRIDE 132   // 128 + 4 pad floats: rows 4 banks apart -> conflict-free b64

// ---------------------------------------------------------------------------
// Kernel 1: fused GEMM (f32 WMMA) + bias + BatchNorm(per-node) + exact GELU
// One wave handles a 16-row x 128-col tile. 10000 % 16 == 0, so a tile never
// crosses a batch boundary (625 tiles per batch image).
// ---------------------------------------------------------------------------
__global__ __launch_bounds__(256) void gcn_gemm_bn_gelu(
    const float* __restrict__ x,      // [B,N,DIN]
    const float* __restrict__ Ww,     // [DOUT,DIN]
    const float* __restrict__ Wb,     // [DOUT]
    const float* __restrict__ gamma,  // [N]
    const float* __restrict__ beta,   // [N]
    const float* __restrict__ mean,   // [N]
    const float* __restrict__ var,    // [N]
    float* __restrict__ h)            // [B,N,DOUT] workspace
{
    __shared__ float Wlds[D_OUT * LDS_W_STRIDE];

    const int tid = threadIdx.x;
    // Cooperative stage of W (64 KB) into padded LDS.
    for (int idx = tid; idx < D_OUT * D_IN; idx += 256) {
        const int o = idx >> 7;
        const int f = idx & 127;
        Wlds[o * LDS_W_STRIDE + f] = Ww[idx];
    }
    __syncthreads();

    const int wave = tid >> 5;
    const int lane = tid & 31;
    const int hsel = lane >> 4;   // half-wave select (K split per ISA layout)
    const int m    = lane & 15;

    const int numTiles = (B_SZ * N_SZ) / 16;        // 2500
    const int tile = blockIdx.x * 8 + wave;
    if (tile >= numTiles) return;                   // wave-uniform branch

    const int row0 = tile * 16;                     // row in [0, B*N)
    const int b    = row0 / N_SZ;

    // --- A fragments: row (row0+m), lanes 0-15 hold K={k,k+1}, 16-31 K={k+2,k+3}
    const v2f* xr = (const v2f*)(x + (size_t)(row0 + m) * D_IN);
    v2f afrag[32];
#pragma unroll
    for (int kk = 0; kk < 32; ++kk)
        afrag[kk] = xr[kk * 2 + hsel];

    // --- Per-row BN constants for the 8 C/D rows this lane-half produces.
    float bn_m[8], bn_s[8], bn_b[8];
#pragma unroll
    for (int j = 0; j < 8; ++j) {
        const int node = row0 - b * N_SZ + j + 8 * hsel;
        const float iv = rsqrtf(var[node] + 1e-5f);
        bn_m[j] = mean[node];
        bn_s[j] = iv * gamma[node];
        bn_b[j] = beta[node];
    }

    // --- 8 column tiles of 16; 32 x V_WMMA_F32_16X16X4_F32 each.
#pragma unroll 1
    for (int c = 0; c < 8; ++c) {
        v8f acc = {};
        const float* wl = &Wlds[(c * 16 + m) * LDS_W_STRIDE + hsel * 2];
#pragma unroll
        for (int kk = 0; kk < 32; ++kk) {
            v2f bfrag = *(const v2f*)(wl + kk * 4);   // ds_load_b64, 8B aligned
            acc = __builtin_amdgcn_wmma_f32_16x16x4_f32(
                false, afrag[kk], false, bfrag, (short)0, acc, false, false);
        }

        const int col = c * 16 + m;
        const float wb = Wb[col];
        float* hp = h + (size_t)row0 * D_OUT + col;
#pragma unroll
        for (int j = 0; j < 8; ++j) {
            float v = acc[j] + wb;
            v = (v - bn_m[j]) * bn_s[j] + bn_b[j];
            // exact GELU: 0.5*v*(1+erf(v/sqrt(2)))
            const float g = 0.5f * v * (1.0f + erff(v * 0.70710678118654752f));
            hp[(size_t)(j + 8 * hsel) * D_OUT] = g;
        }
    }
}

// ---------------------------------------------------------------------------
// Kernel 2: zero the output accumulator (harness poisons d_out with 0xAA).
// ---------------------------------------------------------------------------
__global__ __launch_bounds__(256) void gcn_zero_out(float4* __restrict__ out, int n4)
{
    const int i = blockIdx.x * blockDim.x + threadIdx.x;
    if (i < n4) out[i] = make_float4(0.f, 0.f, 0.f, 0.f);
}

// ---------------------------------------------------------------------------
// Kernel 3: edge scatter. grid = E blocks, 128 thr = 4 waves = 4 batches.
// Edge meta is blockIdx-uniform -> scalar loads. Per lane: one b128 gather
// (512B/wave, coalesced; h lives in 192MB L2) + 4 hardware f32 atomics.
// ---------------------------------------------------------------------------
__global__ __launch_bounds__(128) void gcn_scatter(
    const float* __restrict__ h,      // [B,N,DOUT]
    const int*   __restrict__ ei,     // [2,E]: ei[e]=dst, ei[E+e]=src
    const float* __restrict__ norm,   // [E]
    float* __restrict__ out)          // [B,N,DOUT]
{
    const int e    = blockIdx.x;
    const int bb   = threadIdx.x >> 5;
    const int lane = threadIdx.x & 31;

    const int   dst = ei[e];
    const int   src = ei[E_SZ + e];
    const float w   = norm[e];

    const float4* hp = (const float4*)(h + ((size_t)bb * N_SZ + src) * D_OUT);
    float4 v = hp[lane];
    v.x *= w; v.y *= w; v.z *= w; v.w *= w;

    float* op = out + ((size_t)bb * N_SZ + dst) * D_OUT + lane * 4;
    // Guaranteed hardware FP32 atomic (no return -> STOREcnt), no CAS loop.
    asm volatile("global_atomic_add_f32 %0, %1, off"
                 :: "v"((unsigned long long)(uintptr_t)(op + 0)), "v"(v.x) : "memory");
    asm volatile("global_atomic_add_f32 %0, %1, off"
                 :: "v"((unsigned long long)(uintptr_t)(op + 1)), "v"(v.y) : "memory");
    asm volatile("global_atomic_add_f32 %0, %1, off"
                 :: "v"((unsigned long long)(uintptr_t)(op + 2)), "v"(v.z) : "memory");
    asm volatile("global_atomic_add_f32 %0, %1, off"
                 :: "v"((unsigned long long)(uintptr_t)(op + 3)), "v"(v.w) : "memory");
}

// ---------------------------------------------------------------------------
extern "C" void kernel_launch(void* const* d_in, const int* in_sizes, int n_in,
                              void* d_out, int out_size, void* d_ws, size_t ws_size,
                              hipStream_t stream)
{
    (void)in_sizes; (void)n_in; (void)out_size; (void)ws_size;

    const float* x     = (const float*)d_in[0];
    const int*   ei    = (const int*)  d_in[1];
    const float* norm  = (const float*)d_in[2];
    const float* Ww    = (const float*)d_in[3];
    const float* Wb    = (const float*)d_in[4];
    const float* gamma = (const float*)d_in[5];
    const float* beta  = (const float*)d_in[6];
    const float* mean  = (const float*)d_in[7];
    const float* var   = (const float*)d_in[8];

    float* out = (float*)d_out;
    float* h   = (float*)d_ws;   // B*N*DOUT f32 = 20.48 MB scratch

    // 1) fused GEMM + BN + GELU
    const int numTiles = (B_SZ * N_SZ) / 16;          // 2500 tiles, 8 waves/block
    const int gridA = (numTiles + 7) / 8;             // 313
    gcn_gemm_bn_gelu<<<gridA, 256, 0, stream>>>(x, Ww, Wb, gamma, beta, mean, var, h);

    // 2) zero output accumulator
    const int n4 = (B_SZ * N_SZ * D_OUT) / 4;         // 1.28M float4
    gcn_zero_out<<<(n4 + 255) / 256, 256, 0, stream>>>((float4*)out, n4);

    // 3) edge gather-scale-scatter with hardware f32 atomics
    gcn_scatter<<<E_SZ, 128, 0, stream>>>(h, ei, norm, out);
}